// HierarchicalEmergence_87024627351847
// MI455X (gfx1250) — compile-verified
//
#include <hip/hip_runtime.h>
#include <math.h>

#define NN 1024
#define MM 128
#define KK 16
#define LS 17   // padded LDS row stride (17*4B, coprime with banks)

typedef float v2f __attribute__((ext_vector_type(2)));
typedef float v8f __attribute__((ext_vector_type(8)));

// D(16x16,f32) += A(16x4,f32) * B(4x16,f32)   -> v_wmma_f32_16x16x4_f32
__device__ __forceinline__ v8f wmma4(v2f a, v2f b, v8f c) {
  return __builtin_amdgcn_wmma_f32_16x16x4_f32(
      /*neg_a=*/false, a, /*neg_b=*/false, b,
      /*c_mod=*/(short)0, c, /*reuse_a=*/false, /*reuse_b=*/false);
}

// ---------------------------------------------------------------------------
// Wave-cooperative Cholesky of S (16x16, LDS, stride LS) -> L (lower, LDS).
// Returns logdet(S) (identical value in every lane). Single-wave block.
// ---------------------------------------------------------------------------
__device__ float chol16(const float* S, float* L, int lane) {
  float logdet = 0.f;
  for (int j = 0; j < 16; ++j) {
    // diagonal: redundantly computed by all lanes (broadcast LDS reads)
    float d = S[j * LS + j];
    for (int k = 0; k < j; ++k) { float t = L[j * LS + k]; d -= t * t; }
    float ljj = sqrtf(d);
    logdet += logf(ljj);
    if (lane == 0) L[j * LS + j] = ljj;
    float inv = 1.f / ljj;
    if (lane > j && lane < 16) {
      float v = S[lane * LS + j];
      for (int k = 0; k < j; ++k) v -= L[lane * LS + k] * L[j * LS + k];
      L[lane * LS + j] = v * inv;
    }
    __syncthreads();  // single-wave WG: S_NOP barrier + ds waits
  }
  return 2.f * logdet;
}

// ---------------------------------------------------------------------------
// Kernel 1: invert omega_parent / omega_m_parent (M matrices each, 16x16)
// Gauss-Jordan with partial pivoting; lane = column of the [A | I] tableau.
// grid = (MM, 2), block = 32.
// ---------------------------------------------------------------------------
__global__ __launch_bounds__(32)
void omega_inv_kernel(const float* __restrict__ omega0,
                      const float* __restrict__ omega1,
                      float* __restrict__ inv0, float* __restrict__ inv1) {
  __shared__ float Mx[16 * 33];
  const int m = blockIdx.x;
  const float* src = (blockIdx.y == 0 ? omega0 : omega1) + m * 256;
  float* dst = (blockIdx.y == 0 ? inv0 : inv1) + m * 256;
  const int c = threadIdx.x;  // 0..31

  for (int r = 0; r < 16; ++r) {
    float v = (c < 16) ? src[r * 16 + c] : ((r == (c - 16)) ? 1.f : 0.f);
    Mx[r * 33 + c] = v;
  }
  __syncthreads();

  for (int i = 0; i < 16; ++i) {
    // partial pivot: all lanes scan column i redundantly (broadcast reads)
    int p = i; float best = fabsf(Mx[i * 33 + i]);
    for (int r = i + 1; r < 16; ++r) {
      float v = fabsf(Mx[r * 33 + i]);
      if (v > best) { best = v; p = r; }
    }
    __syncthreads();
    if (p != i) {
      float t1 = Mx[i * 33 + c], t2 = Mx[p * 33 + c];
      Mx[i * 33 + c] = t2; Mx[p * 33 + c] = t1;
    }
    __syncthreads();
    float pinv = 1.f / Mx[i * 33 + i];
    float rowi = Mx[i * 33 + c] * pinv;
    Mx[i * 33 + c] = rowi;
    __syncthreads();
    for (int r = 0; r < 16; ++r) {
      if (r == i) continue;
      float f = Mx[r * 33 + i];          // broadcast read before the write
      Mx[r * 33 + c] -= f * rowi;
    }
    __syncthreads();
  }
  if (c >= 16)
    for (int r = 0; r < 16; ++r) dst[r * 16 + (c - 16)] = Mx[r * 33 + c];
}

// ---------------------------------------------------------------------------
// Kernel 2: Cholesky of sigma_p / sigma_r (N matrices each). Stores L (with
// zeroed upper triangle) and logdet. grid = (NN, 2), block = 32.
// ---------------------------------------------------------------------------
__global__ __launch_bounds__(32)
void chol_pre_kernel(const float* __restrict__ sp, const float* __restrict__ sr,
                     float* __restrict__ Lc0, float* __restrict__ Lc1,
                     float* __restrict__ ld0, float* __restrict__ ld1) {
  __shared__ float S[16 * LS];
  __shared__ float L[16 * LS];
  const int n = blockIdx.x;
  const float* src = (blockIdx.y == 0 ? sp : sr) + n * 256;
  float* Ldst = (blockIdx.y == 0 ? Lc0 : Lc1) + n * 256;
  float* ldet = (blockIdx.y == 0 ? ld0 : ld1);
  const int lane = threadIdx.x;

  for (int idx = lane; idx < 256; idx += 32)
    S[(idx >> 4) * LS + (idx & 15)] = src[idx];
  for (int idx = lane; idx < 16 * LS; idx += 32) L[idx] = 0.f;
  __syncthreads();

  float logdet = chol16(S, L, lane);

  for (int idx = lane; idx < 256; idx += 32)
    Ldst[idx] = L[(idx >> 4) * LS + (idx & 15)];
  if (lane == 0) ldet[n] = logdet;
}

// ---------------------------------------------------------------------------
// Kernel 3: one wave per (n,m) pair; both streams; accumulate W[n,m]*KL.
// grid = NN*MM, block = 32.
// ---------------------------------------------------------------------------
__global__ __launch_bounds__(32)
void kl_pair_kernel(const float* __restrict__ mu_p, const float* __restrict__ omega_child,
                    const float* __restrict__ sigma_q_parent, const float* __restrict__ mu_q_parent,
                    const float* __restrict__ mu_r, const float* __restrict__ omega_m_child,
                    const float* __restrict__ sigma_s_parent, const float* __restrict__ mu_s_parent,
                    const float* __restrict__ W,
                    const float* __restrict__ invA, const float* __restrict__ invB,
                    const float* __restrict__ LcA, const float* __restrict__ LcB,
                    const float* __restrict__ ldA, const float* __restrict__ ldB,
                    float* __restrict__ out) {
  __shared__ float bufT[16 * LS];
  __shared__ float bufU[16 * LS];
  __shared__ float bufS[16 * LS];
  __shared__ float bufL[16 * LS];
  __shared__ float dvec[16];
  __shared__ float tvec[32];

  const int lane = threadIdx.x;
  const int half = lane >> 4;   // which K-pair the lane holds in A/B frags
  const int lid  = lane & 15;   // row (A) / column (B,D) index
  const int bid  = blockIdx.x;
  const int n = bid >> 7;       // / MM
  const int m = bid & (MM - 1);

  float total = 0.f;

  #pragma unroll
  for (int s = 0; s < 2; ++s) {
    const float* mu_c   = (s == 0 ? mu_p : mu_r) + n * KK;
    const float* omg_c  = (s == 0 ? omega_child : omega_m_child) + n * 256;
    const float* oinv   = (s == 0 ? invA : invB) + m * 256;
    const float* sig_p  = (s == 0 ? sigma_q_parent : sigma_s_parent) + m * 256;
    const float* mu_par = (s == 0 ? mu_q_parent : mu_s_parent) + m * KK;
    const float* Lc     = (s == 0 ? LcA : LcB) + n * 256;
    const float  logdet_a = (s == 0 ? ldA : ldB)[n];

    // ---- T = omega_c @ omega_par_inv  (4 chained 16x16x4 WMMAs) ----
    v8f T = {};
    #pragma unroll
    for (int c = 0; c < 4; ++c) {
      const int k0 = 4 * c + 2 * half;
      v2f a, b;
      a[0] = omg_c[lid * 16 + k0];       a[1] = omg_c[lid * 16 + k0 + 1];
      b[0] = oinv[k0 * 16 + lid];        b[1] = oinv[(k0 + 1) * 16 + lid];
      T = wmma4(a, b, T);
    }
    #pragma unroll
    for (int r = 0; r < 8; ++r) bufT[(r + 8 * half) * LS + lid] = T[r];
    __syncthreads();

    // ---- U = T @ sigma_par ----
    v8f U = {};
    #pragma unroll
    for (int c = 0; c < 4; ++c) {
      const int k0 = 4 * c + 2 * half;
      v2f a, b;
      a[0] = bufT[lid * LS + k0];        a[1] = bufT[lid * LS + k0 + 1];
      b[0] = sig_p[k0 * 16 + lid];       b[1] = sig_p[(k0 + 1) * 16 + lid];
      U = wmma4(a, b, U);
    }
    #pragma unroll
    for (int r = 0; r < 8; ++r) bufU[(r + 8 * half) * LS + lid] = U[r];
    __syncthreads();

    // ---- sigma_t = U @ T^T  (B fragment = transposed read of bufT) ----
    v8f Sg = {};
    #pragma unroll
    for (int c = 0; c < 4; ++c) {
      const int k0 = 4 * c + 2 * half;
      v2f a, b;
      a[0] = bufU[lid * LS + k0];        a[1] = bufU[lid * LS + k0 + 1];
      b[0] = bufT[lid * LS + k0];        b[1] = bufT[lid * LS + k0 + 1];  // T^T[k][n] = T[n][k]
      Sg = wmma4(a, b, Sg);
    }
    #pragma unroll
    for (int r = 0; r < 8; ++r) bufS[(r + 8 * half) * LS + lid] = Sg[r];
    __syncthreads();

    // ---- Cholesky: sigma_t = L L^T ----
    const float logdet_b = chol16(bufS, bufL, lane);

    // ---- diff = T @ mu_par - mu_c ----
    {
      float acc = 0.f;
      #pragma unroll
      for (int k = 0; k < 16; ++k) acc += bufT[lid * LS + k] * mu_par[k];
      if (lane < 16) dvec[lane] = acc - mu_c[lane];
    }
    __syncthreads();

    // ---- maha = || L^{-1} diff ||^2 (redundant scalar forward solve) ----
    float w[16];
    float maha = 0.f;
    #pragma unroll
    for (int i = 0; i < 16; ++i) {
      float v = dvec[i];
      #pragma unroll
      for (int j = 0; j < i; ++j) v -= bufL[i * LS + j] * w[j];
      w[i] = v / bufL[i * LS + i];
      maha += w[i] * w[i];
    }

    // ---- trace = || L^{-1} Lc ||_F^2 (lane = column of Lc) ----
    float x[16];
    #pragma unroll
    for (int i = 0; i < 16; ++i) x[i] = (lane < 16) ? Lc[i * 16 + lane] : 0.f;
    #pragma unroll
    for (int j = 0; j < 16; ++j) {
      x[j] /= bufL[j * LS + j];
      #pragma unroll
      for (int i = j + 1; i < 16; ++i) x[i] -= bufL[i * LS + j] * x[j];
    }
    float tl = 0.f;
    #pragma unroll
    for (int i = 0; i < 16; ++i) tl += x[i] * x[i];
    tvec[lane] = tl;
    __syncthreads();
    float trace = 0.f;
    #pragma unroll
    for (int l = 0; l < 16; ++l) trace += tvec[l];

    total += 0.5f * (trace + maha - 16.f + logdet_b - logdet_a);
    __syncthreads();  // buffers reused by the next stream
  }

  if (lane == 0) atomicAdd(out, W[n * MM + m] * total);
}

// ---------------------------------------------------------------------------
extern "C" void kernel_launch(void* const* d_in, const int* in_sizes, int n_in,
                              void* d_out, int out_size, void* d_ws, size_t ws_size,
                              hipStream_t stream) {
  (void)in_sizes; (void)n_in; (void)out_size; (void)ws_size;
  const float* mu_p           = (const float*)d_in[0];
  const float* sigma_p        = (const float*)d_in[1];
  const float* omega_child    = (const float*)d_in[2];
  const float* mu_r           = (const float*)d_in[3];
  const float* sigma_r        = (const float*)d_in[4];
  const float* omega_m_child  = (const float*)d_in[5];
  const float* mu_q_parent    = (const float*)d_in[6];
  const float* sigma_q_parent = (const float*)d_in[7];
  const float* omega_parent   = (const float*)d_in[8];
  const float* mu_s_parent    = (const float*)d_in[9];
  const float* sigma_s_parent = (const float*)d_in[10];
  const float* omega_m_parent = (const float*)d_in[11];
  const float* W              = (const float*)d_in[12];

  float* ws   = (float*)d_ws;
  float* invA = ws;                    // M*256
  float* invB = invA + MM * 256;       // M*256
  float* LcA  = invB + MM * 256;       // N*256
  float* LcB  = LcA + NN * 256;        // N*256
  float* ldA  = LcB + NN * 256;        // N
  float* ldB  = ldA + NN;              // N

  hipMemsetAsync(d_out, 0, sizeof(float), stream);

  omega_inv_kernel<<<dim3(MM, 2), 32, 0, stream>>>(omega_parent, omega_m_parent,
                                                   invA, invB);
  chol_pre_kernel<<<dim3(NN, 2), 32, 0, stream>>>(sigma_p, sigma_r,
                                                  LcA, LcB, ldA, ldB);
  kl_pair_kernel<<<dim3(NN * MM), 32, 0, stream>>>(
      mu_p, omega_child, sigma_q_parent, mu_q_parent,
      mu_r, omega_m_child, sigma_s_parent, mu_s_parent, W,
      invA, invB, LcA, LcB, ldA, ldB, (float*)d_out);
}